// Model_7533372637606
// MI455X (gfx1250) — compile-verified
//
#include <hip/hip_runtime.h>
#include <math.h>

// ---------------------------------------------------------------------------
// Augmented-state Kalman filter (d = 4 + 128) over T=2000 sequential steps.
// Single persistent 1024-thread workgroup; P, W=L^{-1}[HP|r], L live in LDS.
// Structure exploited:
//   predict:   A = blockdiag(F^T, I), F^T is a shear -> row/col adds on P
//   HP[j,:]  = P[sel(j),:] + P[4+j,:]          (H = [OM^T, I], OM^T rows = e0/e1)
//   S[j,k]   = HP[j,sel(k)] + HP[j,4+k] + ov*I
//   W        = L^{-1}[HP | r]   (one blocked forward solve, WMMA trailing updates)
//   K@HP     = W^T W  (SYRK via V_WMMA_F32_16X16X4_F32, symmetric by construction)
//   K@r      = W^T u,  r^T alpha = ||u||^2,  logdet from chol diag
// ---------------------------------------------------------------------------

typedef __attribute__((ext_vector_type(2))) float v2f;
typedef __attribute__((ext_vector_type(8))) float v8f;

#define NT      1024
#define OBSD    128
#define DD      132
#define TSTEPS  2000
#define PSTR    136   // P row stride (floats)
#define WSTR    144   // W row stride: 133 used cols -> 9 tiles of 16
#define LSTR    132   // L row stride

#define P_OFF   0
#define W_OFF   (P_OFF + DD * PSTR)       // 17952
#define L_OFF   (W_OFF + OBSD * WSTR)     // 36384
#define M_OFF   (L_OFF + OBSD * LSTR)     // 53280
#define COL_OFF (M_OFF + DD)              // 53412
#define ACC_OFF (COL_OFF + OBSD)          // 53540
#define SMEM_FLOATS (ACC_OFF + 8)         // 53548 floats = ~214 KB (< 320 KB WGP LDS)

__global__ __launch_bounds__(NT) void kf_kernel(const float* __restrict__ obs,
                                                const float* __restrict__ lbs_p,
                                                const float* __restrict__ lon_p,
                                                const float* __restrict__ ltn_p,
                                                float* __restrict__ out) {
  extern __shared__ float smem[];
  float* Pb   = smem + P_OFF;    // 132 x 132 (stride 136) posterior covariance
  float* Wb   = smem + W_OFF;    // 128 x 133 (stride 144) HP, then W = L^{-1}[HP|r]
  float* Lb   = smem + L_OFF;    // 128 x 128 (stride 132) S, then chol factor L
  float* mv   = smem + M_OFF;    // 132       mean
  float* colv = smem + COL_OFF;  // 128       chol column scratch
  float* accv = smem + ACC_OFF;  // [0]=ll acc, [1]=logdet, [2]=1/diag

  const int tid  = threadIdx.x;
  const int lane = tid & 31;
  const int wave = tid >> 5;
  const int half = lane >> 4;
  const int ml   = lane & 15;

  const float bias_var  = expf(2.0f * lbs_p[0]);
  const float obs_var   = expf(2.0f * lon_p[0]);
  const float trans_var = expf(2.0f * ltn_p[0]);
  const float cll = 117.62413225f;  // 0.5 * 128 * log(2*pi)

  // ---- init: P0 = diag([1e4 x4, bias_var x128]), m0 = 0 ----
  for (int i = tid; i < DD * PSTR; i += NT) Pb[i] = 0.0f;
  for (int i = tid; i < OBSD * WSTR; i += NT) Wb[i] = 0.0f;
  __syncthreads();
  if (tid < DD) {
    Pb[tid * PSTR + tid] = (tid < 4) ? 10000.0f : bias_var;
    mv[tid] = 0.0f;
  }
  if (tid == 0) accv[0] = 0.0f;
  __syncthreads();

#pragma unroll 1
  for (int t = 0; t < TSTEPS; ++t) {
    const float* y = obs + (size_t)t * OBSD;

    // ---- phase 1: predict, left multiply by A (row0+=row2, row1+=row3) ----
    for (int j = tid; j < DD; j += NT) {
      Pb[0 * PSTR + j] += Pb[2 * PSTR + j];
      Pb[1 * PSTR + j] += Pb[3 * PSTR + j];
    }
    __syncthreads();

    // ---- phase 2: right multiply by A^T (col0+=col2, col1+=col3), +Q, m ----
    for (int i = tid; i < DD; i += NT) {
      Pb[i * PSTR + 0] += Pb[i * PSTR + 2];
      Pb[i * PSTR + 1] += Pb[i * PSTR + 3];
      if (i < 4) {  // Q = trans_var * NCV_Q, sparse pattern
        Pb[i * PSTR + i] += trans_var * ((i < 2) ? (1.0f / 3.0f) : 1.0f);
        if (i < 2) Pb[i * PSTR + i + 2] += trans_var * 0.5f;
        else       Pb[i * PSTR + i - 2] += trans_var * 0.5f;
      }
      if (i == 0) { mv[0] += mv[2]; mv[1] += mv[3]; }  // m = A m (shear)
    }
    __syncthreads();

    // ---- phase 3: W <- [HP | r | 0pad];  HP[j,:] = P[sel(j),:] + P[4+j,:] ----
    if (tid == 0) accv[1] = 0.0f;  // logdet accumulator for this step
    for (int j = wave; j < OBSD; j += 32) {
      const int sel = (j < 64) ? 0 : 1;
      const float* Ps = &Pb[sel * PSTR];
      const float* Pj = &Pb[(4 + j) * PSTR];
      for (int c = lane; c < WSTR; c += 32) {
        float v;
        if (c < DD)       v = Ps[c] + Pj[c];
        else if (c == DD) v = y[j] - (mv[sel] + mv[4 + j]);  // r
        else              v = 0.0f;
        Wb[j * WSTR + c] = v;
      }
    }
    __syncthreads();

    // ---- phase 4: S[j,k] = HP[j,sel(k)] + HP[j,4+k] + obs_var*I ----
    for (int j = wave; j < OBSD; j += 32) {
      const float* Wr = &Wb[j * WSTR];
      for (int k = lane; k < OBSD; k += 32) {
        float v = Wr[(k < 64) ? 0 : 1] + Wr[4 + k];
        if (j == k) v += obs_var;
        Lb[j * LSTR + k] = v;
      }
    }
    __syncthreads();

    // ---- phase 5: in-place Cholesky of S (lower), 2 barriers per column ----
#pragma unroll 1
    for (int k = 0; k < OBSD; ++k) {
      for (int i = k + tid; i < OBSD; i += NT) {
        if (i == k) {  // only tid==0
          float d = sqrtf(Lb[k * LSTR + k]);
          Lb[k * LSTR + k] = d;
          accv[2] = 1.0f / d;
          accv[1] += logf(d);
        } else {
          colv[i] = Lb[i * LSTR + k];
        }
      }
      __syncthreads();
      const float inv  = accv[2];
      const float inv2 = inv * inv;
      for (int i = k + 1 + tid; i < OBSD; i += NT) Lb[i * LSTR + k] = colv[i] * inv;
      for (int i = k + 1 + wave; i < OBSD; i += 32) {
        const float ci = colv[i] * inv2;
        for (int j = k + 1 + lane; j <= i; j += 32) Lb[i * LSTR + j] -= ci * colv[j];
      }
      __syncthreads();
    }

    // ---- phase 6: blocked forward solve  W = L^{-1} [HP | r] ----
#pragma unroll 1
    for (int b = 0; b < 8; ++b) {
      // trailing update via WMMA f32 16x16x4 chains: W_b -= sum_{kb<b} L[b,kb] W_kb
      if (b > 0 && wave < 9) {
        const int cj = wave * 16;  // column tile (9 tiles cover 144 cols incl. r)
        v8f acc8 = {0.f, 0.f, 0.f, 0.f, 0.f, 0.f, 0.f, 0.f};
#pragma unroll 1
        for (int kb = 0; kb < b; ++kb) {
#pragma unroll
          for (int q = 0; q < 4; ++q) {
            const int kc = kb * 16 + q * 4;
            v2f av, bv;
            av.x = Lb[(b * 16 + ml) * LSTR + kc + 2 * half];
            av.y = Lb[(b * 16 + ml) * LSTR + kc + 2 * half + 1];
            bv.x = Wb[(kc + 2 * half) * WSTR + cj + ml];
            bv.y = Wb[(kc + 2 * half + 1) * WSTR + cj + ml];
            acc8 = __builtin_amdgcn_wmma_f32_16x16x4_f32(
                false, av, false, bv, (short)0, acc8, false, false);
          }
        }
#pragma unroll
        for (int v = 0; v < 8; ++v) {
          const int row = b * 16 + v + 8 * half;
          Wb[row * WSTR + cj + ml] -= acc8[v];
        }
      }
      __syncthreads();
      // diagonal 16x16 triangular solve, one thread per RHS column
      if (tid < WSTR) {
        const int c = tid;
        float w[16];
#pragma unroll
        for (int i = 0; i < 16; ++i) w[i] = Wb[(b * 16 + i) * WSTR + c];
#pragma unroll
        for (int i = 0; i < 16; ++i) {
          float s = w[i];
#pragma unroll
          for (int k2 = 0; k2 < i; ++k2) s -= Lb[(b * 16 + i) * LSTR + b * 16 + k2] * w[k2];
          s /= Lb[(b * 16 + i) * LSTR + b * 16 + i];
          w[i] = s;
          Wb[(b * 16 + i) * WSTR + c] = s;
        }
      }
      __syncthreads();
    }

    // ---- phase 7: m += W^T u ;  ll += -0.5||u||^2 - logdet - const ----
    if (tid < DD) {
      float s = 0.0f;
      for (int r = 0; r < OBSD; ++r) s += Wb[r * WSTR + tid] * Wb[r * WSTR + DD];
      mv[tid] += s;
    } else if (tid == DD) {
      float s = 0.0f;
      for (int r = 0; r < OBSD; ++r) { const float u = Wb[r * WSTR + DD]; s += u * u; }
      accv[0] += -0.5f * s - accv[1] - cll;
    }
    __syncthreads();

    // ---- phase 8: SYRK  P -= W^T W  (81 tiles of 16x16, K=128, WMMA) ----
#pragma unroll 1
    for (int tile = wave; tile < 81; tile += 32) {
      const int ti = tile / 9, tj = tile - ti * 9;
      v8f acc8 = {0.f, 0.f, 0.f, 0.f, 0.f, 0.f, 0.f, 0.f};
#pragma unroll 1
      for (int kb = 0; kb < 8; ++kb) {
#pragma unroll
        for (int q = 0; q < 4; ++q) {
          const int r0 = kb * 16 + q * 4 + 2 * half;
          v2f av, bv;
          av.x = Wb[r0 * WSTR + ti * 16 + ml];        // A[M][K] = W[r][ti*16+M]
          av.y = Wb[(r0 + 1) * WSTR + ti * 16 + ml];
          bv.x = Wb[r0 * WSTR + tj * 16 + ml];        // B[K][N] = W[r][tj*16+N]
          bv.y = Wb[(r0 + 1) * WSTR + tj * 16 + ml];
          acc8 = __builtin_amdgcn_wmma_f32_16x16x4_f32(
              false, av, false, bv, (short)0, acc8, false, false);
        }
      }
#pragma unroll
      for (int v = 0; v < 8; ++v) {
        const int row = ti * 16 + v + 8 * half;
        const int col = tj * 16 + ml;
        if (row < DD && col < DD) Pb[row * PSTR + col] -= acc8[v];
      }
    }
    __syncthreads();
  }

  // ---- outputs: [logp, m[0:4], P[0:4,0:4] row-major] = 21 floats ----
  if (tid == 0) out[0] = accv[0];
  if (tid < 4) out[1 + tid] = mv[tid];
  if (tid < 16) out[5 + tid] = Pb[(tid >> 2) * PSTR + (tid & 3)];
}

extern "C" void kernel_launch(void* const* d_in, const int* in_sizes, int n_in,
                              void* d_out, int out_size, void* d_ws, size_t ws_size,
                              hipStream_t stream) {
  (void)in_sizes; (void)n_in; (void)d_ws; (void)ws_size; (void)out_size;
  const float* obs = (const float*)d_in[0];
  const float* lbs = (const float*)d_in[1];
  const float* lon = (const float*)d_in[2];
  const float* ltn = (const float*)d_in[3];
  float* out = (float*)d_out;

  const size_t smem = (size_t)SMEM_FLOATS * sizeof(float);
  hipFuncSetAttribute(reinterpret_cast<const void*>(kf_kernel),
                      hipFuncAttributeMaxDynamicSharedMemorySize, (int)smem);
  kf_kernel<<<dim3(1), dim3(NT), smem, stream>>>(obs, lbs, lon, ltn, out);
}